// AttentionDecoder_64665027608694
// MI455X (gfx1250) — compile-verified
//
#include <hip/hip_runtime.h>
#include <hip/hip_bf16.h>

// AttentionDecoder for MI455X (gfx1250).
// Bandwidth-bound on time_matrix_K/V (2 x 268MB per block) -> coalesced f32
// float4 streaming for those; all GEMM-shaped math uses V_WMMA_F32_16X16X4_F32
// with 4x register blocking along M (one B fragment feeds 4 WMMAs) and fully
// constant-folded addressing (all leading dims are compile-time 256).

#define Bdim    4
#define Ldim    256
#define HD      256
#define NHEADS  4
#define Dh      64
#define NBLOCKS 2
#define EPSV    1e-8f
#define NEGV    (-4294967295.0f)

typedef __attribute__((ext_vector_type(2))) float v2f;
typedef __attribute__((ext_vector_type(8))) float v8f;

// ---------------- RMSNorm: one block per (b,l) row, 256 threads ----------------
__global__ void rmsnorm_kernel(const float* __restrict__ x, const float* __restrict__ w,
                               float* __restrict__ y) {
  __shared__ float red[8];
  long row = blockIdx.x;
  int t = threadIdx.x;
  float v = x[row * HD + t];
  float ss = v * v;
#pragma unroll
  for (int o = 16; o > 0; o >>= 1) ss += __shfl_xor(ss, o, 32);
  if ((t & 31) == 0) red[t >> 5] = ss;
  __syncthreads();
  float tot = 0.f;
#pragma unroll
  for (int j = 0; j < 8; ++j) tot += red[j];
  float sc = rsqrtf(tot * (1.0f / HD) + EPSV);
  y[row * HD + t] = v * sc * w[t];
}

// ---------------- Batched WMMA f32 GEMM, 64x16 strip per wave ----------------
// C[m,n] = sum_k A[m,k]*B[k,n] (+bias[n]) (+E[m,n]) (relu?)
// A element (m,k) at A + m*256 + k ; C element (m,n) at C + m*256 + n
// B element (k,n) at Bm + k*LDBK + n*LDBN  (LDBK=1,LDBN=256: B=W^T ; LDBK=256,LDBN=1: plain B)
// batch z = z1*B2 + z2 ; per-operand offsets z1*s?1 + z2*s?2
// Wave computes 4 vertically-stacked 16x16 tiles: each B fragment -> 4 WMMAs.
template <int LDBK, int LDBN, int KDIM>
__global__ void wmma_gemm_kernel(const float* __restrict__ A, const float* __restrict__ Bm,
                                 const float* __restrict__ bias, const float* __restrict__ E,
                                 float* __restrict__ C,
                                 int B2,
                                 long long sA1, long long sA2,
                                 long long sB1, long long sB2,
                                 long long sC1, long long sC2,
                                 int relu) {
  constexpr int LD = 256;  // lda == ldc == 256 for every GEMM in this model
  int z  = blockIdx.z;
  int z1 = z / B2, z2 = z % B2;
  A  += z1 * sA1 + z2 * sA2;
  Bm += z1 * sB1 + z2 * sB2;
  C  += z1 * sC1 + z2 * sC2;

  int wave = threadIdx.x >> 5;
  int lane = threadIdx.x & 31;
  int m0 = blockIdx.x * 64;                 // 4 M-tiles per wave
  int n0 = (blockIdx.y * 4 + wave) * 16;

  int half = lane >> 4;   // 0: K=k0+0/1 ; 1: K=k0+2/3  (ISA A/B fragment layout)
  int l16  = lane & 15;

  const float* ap = A  + (long)(m0 + l16) * LD + 2 * half;
  const float* bp = Bm + (long)(n0 + l16) * LDBN + (long)(2 * half) * LDBK;

  v8f acc[4] = {};
#pragma unroll 8
  for (int k0 = 0; k0 < KDIM; k0 += 4) {
    v2f b;
    b.x = bp[(long)k0 * LDBK];
    b.y = bp[(long)k0 * LDBK + LDBK];
#pragma unroll
    for (int t = 0; t < 4; ++t) {
      v2f a;
      a.x = ap[t * 16 * LD + k0];
      a.y = ap[t * 16 * LD + k0 + 1];
      acc[t] = __builtin_amdgcn_wmma_f32_16x16x4_f32(false, a, false, b,
                                                     (short)0, acc[t], false, false);
    }
  }
#pragma unroll
  for (int t = 0; t < 4; ++t) {
#pragma unroll
    for (int r = 0; r < 8; ++r) {
      int mm = m0 + t * 16 + r + 8 * half;  // C/D layout: lanes 0-15 M=r, lanes 16-31 M=r+8
      int nn = n0 + l16;
      float v = acc[t][r];
      if (bias) v += bias[nn];
      if (E)    v += E[(long)mm * LD + nn];
      if (relu) v = v > 0.f ? v : 0.f;
      C[(long)mm * LD + nn] = v;
    }
  }
}

// ---------------- tmK dot + mask + scale (in place on S) ----------------
// One block per (b,q); 8 waves; wave handles k = t*8+wave; lane covers 8
// contiguous H columns -> 1KB fully-coalesced row reads of tmK[b,q,k,:].
__global__ void scores_tm_mask_kernel(const float* __restrict__ tmK,
                                      const float* __restrict__ Q,
                                      float* __restrict__ S,
                                      const unsigned char* __restrict__ tl,
                                      float scale) {
  int bq = blockIdx.x;
  int b = bq >> 8;
  int q = bq & (Ldim - 1);
  int wave = threadIdx.x >> 5;
  int lane = threadIdx.x & 31;
  const float* qrow = Q + (long)bq * HD + lane * 8;
  float4 q0 = *(const float4*)qrow;
  float4 q1 = *(const float4*)(qrow + 4);
  int head = lane >> 3;                 // 8 lanes per head (64 cols)
  bool rowmask = tl[bq] != 0;
  const float* tbase = tmK + (long)bq * Ldim * HD;
  for (int t = 0; t < Ldim / 8; ++t) {
    int k = t * 8 + wave;
    const float* r = tbase + (long)k * HD + lane * 8;
    __builtin_prefetch(r + 8 * HD, 0, 0);
    float4 a0 = *(const float4*)r;
    float4 a1 = *(const float4*)(r + 4);
    float s = a0.x * q0.x + a0.y * q0.y + a0.z * q0.z + a0.w * q0.w +
              a1.x * q1.x + a1.y * q1.y + a1.z * q1.z + a1.w * q1.w;
    s += __shfl_xor(s, 1, 32);
    s += __shfl_xor(s, 2, 32);
    s += __shfl_xor(s, 4, 32);
    if ((lane & 7) == 0) {
      long idx = (((long)head * Bdim + b) * Ldim + q) * Ldim + k;
      float val = S[idx] + s;
      S[idx] = (rowmask || (k > q)) ? NEGV : val * scale;
    }
  }
}

// ---------------- softmax: one wave per row of 256 ----------------
__global__ void softmax_kernel(float* __restrict__ S) {
  long row = (long)blockIdx.x * 8 + (threadIdx.x >> 5);
  int lane = threadIdx.x & 31;
  float* p = S + row * Ldim + lane;
  float v[8];
  float mx = -INFINITY;
#pragma unroll
  for (int i = 0; i < 8; ++i) { v[i] = p[i * 32]; mx = fmaxf(mx, v[i]); }
#pragma unroll
  for (int o = 16; o > 0; o >>= 1) mx = fmaxf(mx, __shfl_xor(mx, o, 32));
  float sum = 0.f;
#pragma unroll
  for (int i = 0; i < 8; ++i) { v[i] = __expf(v[i] - mx); sum += v[i]; }
#pragma unroll
  for (int o = 16; o > 0; o >>= 1) sum += __shfl_xor(sum, o, 32);
  float inv = 1.0f / sum;
#pragma unroll
  for (int i = 0; i < 8; ++i) p[i * 32] = v[i] * inv;
}

// ---------------- tmV contraction: out[b,q,c] += sum_k P[h(c),b,q,k]*tmV[b,q,k,c] ----------------
// One block per (b,q); probs rows (4 heads) cached in LDS; tmV rows read 1KB coalesced.
__global__ void out_tm_kernel(const float* __restrict__ tmV,
                              const float* __restrict__ P,
                              float* __restrict__ out) {
  __shared__ float sa[NHEADS * Ldim];
  int bq = blockIdx.x;
  int b = bq >> 8;
  int q = bq & (Ldim - 1);
  int t = threadIdx.x;
#pragma unroll
  for (int h = 0; h < NHEADS; ++h)
    sa[h * Ldim + t] = P[(((long)h * Bdim + b) * Ldim + q) * Ldim + t];
  __syncthreads();
  const float* sarow = sa + (t >> 6) * Ldim;      // head = column/64
  const float* tv = tmV + (long)bq * Ldim * HD + t;
  float acc = 0.f;
#pragma unroll 4
  for (int k = 0; k < Ldim; ++k) acc += sarow[k] * tv[(long)k * HD];
  out[(long)bq * HD + t] += acc;
}

// ---------------- elementwise ----------------
__global__ void add_kernel(float* __restrict__ a, const float* __restrict__ b) {
  long i = (long)blockIdx.x * blockDim.x + threadIdx.x;
  a[i] += b[i];
}

__global__ void add_notpad_kernel(float* __restrict__ a, const float* __restrict__ b,
                                  const unsigned char* __restrict__ tl) {
  long i = (long)blockIdx.x * blockDim.x + threadIdx.x;
  float np = tl[i / HD] ? 0.0f : 1.0f;
  a[i] = (a[i] + b[i]) * np;
}

extern "C" void kernel_launch(void* const* d_in, const int* in_sizes, int n_in,
                              void* d_out, int out_size, void* d_ws, size_t ws_size,
                              hipStream_t stream) {
  (void)in_sizes; (void)n_in; (void)out_size; (void)ws_size;
  const float* seqs   = (const float*)d_in[0];
  const unsigned char* tl = (const unsigned char*)d_in[2];   // timeline_mask (bool)
  const float* tmK    = (const float*)d_in[3];
  const float* tmV    = (const float*)d_in[4];
  const float* apK    = (const float*)d_in[5];
  const float* apV    = (const float*)d_in[6];
  const float* ln_attn= (const float*)d_in[7];
  const float* Qw = (const float*)d_in[8];
  const float* Qb = (const float*)d_in[9];
  const float* Kw = (const float*)d_in[10];
  const float* Kb = (const float*)d_in[11];
  const float* Vw = (const float*)d_in[12];
  const float* Vb = (const float*)d_in[13];
  const float* ln_ffn = (const float*)d_in[14];
  const float* W1 = (const float*)d_in[15];
  const float* b1 = (const float*)d_in[16];
  const float* W2 = (const float*)d_in[17];
  const float* b2 = (const float*)d_in[18];
  const float* ln_last = (const float*)d_in[19];
  float* out = (float*)d_out;

  const long BLH = (long)Bdim * Ldim * HD;               // 262144
  float* ws   = (float*)d_ws;
  float* cur  = ws;
  float* xn   = cur  + BLH;
  float* bufQ = xn   + BLH;
  float* bufK = bufQ + BLH;   // K + apK
  float* bufV = bufK + BLH;   // V + apV
  float* bufO = bufV + BLH;   // attention output (merged heads)
  float* bufS = bufO + BLH;   // scores/probs [h,B,L,L] (4MB)

  hipMemcpyAsync(cur, seqs, BLH * sizeof(float), hipMemcpyDeviceToDevice, stream);

  const float scale = 0.125f;   // 1/sqrt(64)
  dim3 gblk(128, 1, 1);
  const long long LH = (long long)Ldim * HD;
  const long long LL = (long long)Ldim * Ldim;

  for (int i = 0; i < NBLOCKS; ++i) {
    // x = rmsnorm(seqs)
    rmsnorm_kernel<<<Bdim * Ldim, HD, 0, stream>>>(cur, ln_attn + i * HD, xn);
    // Q = x @ Qw^T + Qb              [M=1024, N=256, K=256]
    wmma_gemm_kernel<1, 256, 256><<<dim3(16, 4, 1), gblk, 0, stream>>>(
        xn, Qw + (long)i * HD * HD, Qb + i * HD, nullptr, bufQ,
        1, 0, 0, 0, 0, 0, 0, 0);
    // K2 = seqs @ Kw^T + Kb + abs_pos_K
    wmma_gemm_kernel<1, 256, 256><<<dim3(16, 4, 1), gblk, 0, stream>>>(
        cur, Kw + (long)i * HD * HD, Kb + i * HD, apK, bufK,
        1, 0, 0, 0, 0, 0, 0, 0);
    // V2 = seqs @ Vw^T + Vb + abs_pos_V
    wmma_gemm_kernel<1, 256, 256><<<dim3(16, 4, 1), gblk, 0, stream>>>(
        cur, Vw + (long)i * HD * HD, Vb + i * HD, apV, bufV,
        1, 0, 0, 0, 0, 0, 0, 0);
    // S = Q @ K2^T per (h,b)         [M=256, N=256, K=64, batch=16]
    wmma_gemm_kernel<1, 256, 64><<<dim3(4, 4, 16), gblk, 0, stream>>>(
        bufQ, bufK, nullptr, nullptr, bufS,
        Bdim, Dh, LH, Dh, LH, (long long)Bdim * LL, LL, 0);
    // S += tmK . Q ; mask ; scale    (streams 268MB of tmK, coalesced)
    scores_tm_mask_kernel<<<Bdim * Ldim, 256, 0, stream>>>(tmK, bufQ, bufS, tl, scale);
    // softmax rows
    softmax_kernel<<<(NHEADS * Bdim * Ldim) / 8, 256, 0, stream>>>(bufS);
    // out = A @ V2 per (h,b)         [M=256, N=64, K=256, batch=16]
    wmma_gemm_kernel<256, 1, 256><<<dim3(4, 1, 16), gblk, 0, stream>>>(
        bufS, bufV, nullptr, nullptr, bufO,
        Bdim, (long long)Bdim * LL, LL, Dh, LH, Dh, LH, 0);
    // out += A . tmV                 (streams 268MB of tmV, coalesced)
    out_tm_kernel<<<Bdim * Ldim, 256, 0, stream>>>(tmV, bufS, bufO);
    // residual
    add_kernel<<<(int)(BLH / 256), 256, 0, stream>>>(cur, bufO);
    // FFN
    rmsnorm_kernel<<<Bdim * Ldim, HD, 0, stream>>>(cur, ln_ffn + i * HD, xn);
    wmma_gemm_kernel<1, 256, 256><<<dim3(16, 4, 1), gblk, 0, stream>>>(
        xn, W1 + (long)i * HD * HD, b1 + i * HD, nullptr, bufQ,
        1, 0, 0, 0, 0, 0, 0, 1 /*relu*/);
    wmma_gemm_kernel<1, 256, 256><<<dim3(16, 4, 1), gblk, 0, stream>>>(
        bufQ, W2 + (long)i * HD * HD, b2 + i * HD, nullptr, bufK,
        1, 0, 0, 0, 0, 0, 0, 0);
    add_notpad_kernel<<<(int)(BLH / 256), 256, 0, stream>>>(cur, bufK, tl);
  }
  rmsnorm_kernel<<<Bdim * Ldim, HD, 0, stream>>>(cur, ln_last, out);
}